// SpatialGNNEncoder_14851996909776
// MI455X (gfx1250) — compile-verified
//
#include <hip/hip_runtime.h>
#include <math.h>

// Problem constants (match reference)
#define N_NODES 100000
#define E_EDGES 1600000
#define ETOT (E_EDGES + N_NODES)   // reference appends N self-loops
#define M_IN 40
#define HID 128
#define HEADS 4
#define DH 32
#define LAT 64
#define NEG_SLOPE 0.2f
#define LN_EPS 1e-5f

typedef float v2f __attribute__((ext_vector_type(2)));
typedef float v8f __attribute__((ext_vector_type(8)));

static __device__ __forceinline__ float warp_sum32(float v) {
#pragma unroll
  for (int off = 16; off > 0; off >>= 1) v += __shfl_xor(v, off, 32);
  return v;
}

static __device__ __forceinline__ float lrelu(float t) {
  return t > 0.0f ? t : t * NEG_SLOPE;
}

static __device__ __forceinline__ float gelu_exact(float t) {
  return 0.5f * t * (1.0f + erff(t * 0.70710678118654752f));
}

// ---------------------------------------------------------------------------
// init: agg = 0, m = -FLT_MAX, denom = 0  (must rerun every call: harness does
// not re-poison between replays and atomics accumulate into these buffers)
// ---------------------------------------------------------------------------
__global__ void init_kernel(float* __restrict__ agg, float* __restrict__ m,
                            float* __restrict__ denom) {
  int idx = blockIdx.x * blockDim.x + threadIdx.x;
  if (idx < N_NODES * HID) agg[idx] = 0.0f;
  if (idx < N_NODES * HEADS) {
    m[idx] = -3.402823466e+38f;
    denom[idx] = 0.0f;
  }
}

// ---------------------------------------------------------------------------
// h = x @ Wp + bp   (N x 40) @ (40 x 128), WMMA f32 16x16x4, K-steps = 10.
// Block = 256 threads = 8 waves; wave w owns column tile w; blockIdx = row tile.
// ---------------------------------------------------------------------------
__global__ __launch_bounds__(256) void proj_kernel(
    const float* __restrict__ x, const float* __restrict__ Wp,
    const float* __restrict__ bp, float* __restrict__ h) {
  __shared__ float lx[16 * M_IN];
  const int tile_m = blockIdx.x;
  const int tid = threadIdx.x;
  for (int i = tid; i < 16 * M_IN; i += 256)
    lx[i] = x[(tile_m * 16 + i / M_IN) * M_IN + (i % M_IN)];
  __syncthreads();

  const int wave = tid >> 5, lane = tid & 31;
  const int half = lane >> 4, mrow = lane & 15;
  const int col = wave * 16 + mrow;
  v8f c = {};
#pragma unroll
  for (int k0 = 0; k0 < M_IN; k0 += 4) {
    v2f a, b;
    a.x = lx[mrow * M_IN + k0 + 2 * half];
    a.y = lx[mrow * M_IN + k0 + 2 * half + 1];
    b.x = Wp[(k0 + 2 * half) * HID + col];
    b.y = Wp[(k0 + 2 * half + 1) * HID + col];
    c = __builtin_amdgcn_wmma_f32_16x16x4_f32(false, a, false, b, (short)0, c,
                                              false, false);
  }
  const float bias = bp[col];
#pragma unroll
  for (int r = 0; r < 8; ++r) {
    int row = tile_m * 16 + r + 8 * half;
    h[row * HID + col] = c[r] + bias;
  }
}

// ---------------------------------------------------------------------------
// hl = h @ Wl ; hr = h @ Wr  fused (A tile + A fragments shared), K = 128.
// ---------------------------------------------------------------------------
__global__ __launch_bounds__(256) void dual_gemm_kernel(
    const float* __restrict__ h, const float* __restrict__ Wl,
    const float* __restrict__ Wr, float* __restrict__ hl,
    float* __restrict__ hr) {
  __shared__ float la[16 * HID];
  const int tile_m = blockIdx.x;
  const int tid = threadIdx.x;
  for (int i = tid; i < 16 * HID; i += 256) la[i] = h[tile_m * 16 * HID + i];
  __syncthreads();

  const int wave = tid >> 5, lane = tid & 31;
  const int half = lane >> 4, mrow = lane & 15;
  const int col = wave * 16 + mrow;
  v8f cl = {}, cr = {};
#pragma unroll 4
  for (int k0 = 0; k0 < HID; k0 += 4) {
    v2f a, bl, br;
    a.x = la[mrow * HID + k0 + 2 * half];
    a.y = la[mrow * HID + k0 + 2 * half + 1];
    bl.x = Wl[(k0 + 2 * half) * HID + col];
    bl.y = Wl[(k0 + 2 * half + 1) * HID + col];
    br.x = Wr[(k0 + 2 * half) * HID + col];
    br.y = Wr[(k0 + 2 * half + 1) * HID + col];
    cl = __builtin_amdgcn_wmma_f32_16x16x4_f32(false, a, false, bl, (short)0,
                                               cl, false, false);
    cr = __builtin_amdgcn_wmma_f32_16x16x4_f32(false, a, false, br, (short)0,
                                               cr, false, false);
  }
#pragma unroll
  for (int r = 0; r < 8; ++r) {
    int row = tile_m * 16 + r + 8 * half;
    hl[row * HID + col] = cl[r];
    hr[row * HID + col] = cr[r];
  }
}

// ---------------------------------------------------------------------------
// out = h @ Wo + bo  (N x 128) @ (128 x 64): 4 col tiles -> 4 waves / block.
// ---------------------------------------------------------------------------
__global__ __launch_bounds__(128) void out_gemm_kernel(
    const float* __restrict__ h, const float* __restrict__ Wo,
    const float* __restrict__ bo, float* __restrict__ out) {
  __shared__ float la[16 * HID];
  const int tile_m = blockIdx.x;
  const int tid = threadIdx.x;
  for (int i = tid; i < 16 * HID; i += 128) la[i] = h[tile_m * 16 * HID + i];
  __syncthreads();

  const int wave = tid >> 5, lane = tid & 31;
  const int half = lane >> 4, mrow = lane & 15;
  const int col = wave * 16 + mrow;
  v8f c = {};
#pragma unroll 4
  for (int k0 = 0; k0 < HID; k0 += 4) {
    v2f a, b;
    a.x = la[mrow * HID + k0 + 2 * half];
    a.y = la[mrow * HID + k0 + 2 * half + 1];
    b.x = Wo[(k0 + 2 * half) * LAT + col];
    b.y = Wo[(k0 + 2 * half + 1) * LAT + col];
    c = __builtin_amdgcn_wmma_f32_16x16x4_f32(false, a, false, b, (short)0, c,
                                              false, false);
  }
  const float bias = bo[col];
#pragma unroll
  for (int r = 0; r < 8; ++r) {
    int row = tile_m * 16 + r + 8 * half;
    out[row * LAT + col] = c[r] + bias;
  }
}

// ---------------------------------------------------------------------------
// Warp-per-edge attention logits + segment max (native f32 max atomic:
// GLOBAL_ATOMIC_MAX_NUM_F32, non-returning -> STOREcnt).
// Lane l covers flat dims [4l, 4l+4) of the 128-wide row; head = l >> 3.
// ---------------------------------------------------------------------------
__global__ __launch_bounds__(256) void edge_logits_kernel(
    const int* __restrict__ ei, const float* __restrict__ hl,
    const float* __restrict__ hr, const float* __restrict__ att,
    float* __restrict__ logits, float* __restrict__ m) {
  const int e = blockIdx.x * 8 + (threadIdx.x >> 5);
  if (e >= ETOT) return;
  const int lane = threadIdx.x & 31;
  int s, d;
  if (e < E_EDGES) {
    s = ei[e];
    d = ei[E_EDGES + e];
  } else {
    s = d = e - E_EDGES;
  }
  const float4 a = *(const float4*)(hl + s * HID + lane * 4);
  const float4 b = *(const float4*)(hr + d * HID + lane * 4);
  const float4 w = *(const float4*)(att + lane * 4);
  float dot = lrelu(a.x + b.x) * w.x + lrelu(a.y + b.y) * w.y +
              lrelu(a.z + b.z) * w.z + lrelu(a.w + b.w) * w.w;
  dot += __shfl_xor(dot, 1, 32);
  dot += __shfl_xor(dot, 2, 32);
  dot += __shfl_xor(dot, 4, 32);
  if ((lane & 7) == 0) {
    const int head = lane >> 3;
    logits[e * HEADS + head] = dot;
    unsafeAtomicMax(&m[d * HEADS + head], dot);
  }
}

// ---------------------------------------------------------------------------
// ex = exp(logit - m[dst]); denom[dst] += ex  (thread per edge-head,
// native GLOBAL_ATOMIC_ADD_F32)
// ---------------------------------------------------------------------------
__global__ void edge_exp_kernel(const int* __restrict__ ei,
                                float* __restrict__ logits,
                                const float* __restrict__ m,
                                float* __restrict__ denom) {
  const int idx = blockIdx.x * blockDim.x + threadIdx.x;
  if (idx >= ETOT * HEADS) return;
  const int e = idx >> 2;
  const int head = idx & 3;
  const int d = (e < E_EDGES) ? ei[E_EDGES + e] : (e - E_EDGES);
  const float ex = __expf(logits[idx] - m[d * HEADS + head]);
  logits[idx] = ex;
  unsafeAtomicAdd(&denom[d * HEADS + head], ex);
}

// ---------------------------------------------------------------------------
// agg[dst] += alpha * hl[src]   (warp per edge, 4 native f32 atomics per lane)
// ---------------------------------------------------------------------------
__global__ __launch_bounds__(256) void edge_scatter_kernel(
    const int* __restrict__ ei, const float* __restrict__ hl,
    const float* __restrict__ logits, const float* __restrict__ denom,
    float* __restrict__ agg) {
  const int e = blockIdx.x * 8 + (threadIdx.x >> 5);
  if (e >= ETOT) return;
  const int lane = threadIdx.x & 31;
  int s, d;
  if (e < E_EDGES) {
    s = ei[e];
    d = ei[E_EDGES + e];
  } else {
    s = d = e - E_EDGES;
  }
  const int head = lane >> 3;
  const float alpha = logits[e * HEADS + head] / denom[d * HEADS + head];
  const float4 hv = *(const float4*)(hl + s * HID + lane * 4);
  float* o = agg + d * HID + lane * 4;
  unsafeAtomicAdd(o + 0, hv.x * alpha);
  unsafeAtomicAdd(o + 1, hv.y * alpha);
  unsafeAtomicAdd(o + 2, hv.z * alpha);
  unsafeAtomicAdd(o + 3, hv.w * alpha);
}

// ---------------------------------------------------------------------------
// hout = LayerNorm(gelu(agg + bias) + hin) * g + be    (warp per node)
// ---------------------------------------------------------------------------
__global__ __launch_bounds__(256) void post_kernel(
    const float* __restrict__ agg, const float* __restrict__ bias,
    const float* __restrict__ hin, const float* __restrict__ g,
    const float* __restrict__ be, float* __restrict__ hout) {
  const int node = blockIdx.x * 8 + (threadIdx.x >> 5);
  if (node >= N_NODES) return;
  const int lane = threadIdx.x & 31;
  const float4 a = *(const float4*)(agg + node * HID + lane * 4);
  const float4 b4 = *(const float4*)(bias + lane * 4);
  const float4 hv = *(const float4*)(hin + node * HID + lane * 4);
  float x0 = gelu_exact(a.x + b4.x) + hv.x;
  float x1 = gelu_exact(a.y + b4.y) + hv.y;
  float x2 = gelu_exact(a.z + b4.z) + hv.z;
  float x3 = gelu_exact(a.w + b4.w) + hv.w;
  const float mu = warp_sum32(x0 + x1 + x2 + x3) * (1.0f / HID);
  const float d0 = x0 - mu, d1 = x1 - mu, d2 = x2 - mu, d3 = x3 - mu;
  const float var =
      warp_sum32(d0 * d0 + d1 * d1 + d2 * d2 + d3 * d3) * (1.0f / HID);
  const float inv = rsqrtf(var + LN_EPS);
  const float4 g4 = *(const float4*)(g + lane * 4);
  const float4 be4 = *(const float4*)(be + lane * 4);
  float4 o;
  o.x = d0 * inv * g4.x + be4.x;
  o.y = d1 * inv * g4.y + be4.y;
  o.z = d2 * inv * g4.z + be4.z;
  o.w = d3 * inv * g4.w + be4.w;
  *(float4*)(hout + node * HID + lane * 4) = o;
}

// ---------------------------------------------------------------------------
extern "C" void kernel_launch(void* const* d_in, const int* in_sizes, int n_in,
                              void* d_out, int out_size, void* d_ws,
                              size_t ws_size, hipStream_t stream) {
  const float* x = (const float*)d_in[0];
  const int* ei = (const int*)d_in[1];  // [2, E] (src row then dst row)
  const float* Wp = (const float*)d_in[2];
  const float* bp = (const float*)d_in[3];
  const float* W1l = (const float*)d_in[4];
  const float* W1r = (const float*)d_in[5];
  const float* att1 = (const float*)d_in[6];
  const float* b1 = (const float*)d_in[7];
  const float* W2l = (const float*)d_in[8];
  const float* W2r = (const float*)d_in[9];
  const float* att2 = (const float*)d_in[10];
  const float* b2 = (const float*)d_in[11];
  const float* g1 = (const float*)d_in[12];
  const float* be1 = (const float*)d_in[13];
  const float* g2 = (const float*)d_in[14];
  const float* be2 = (const float*)d_in[15];
  const float* Wo = (const float*)d_in[16];
  const float* bo = (const float*)d_in[17];
  float* out = (float*)d_out;

  // Workspace layout (floats): 4 N*HID ping-pong buffers + logits + m + denom
  // total = 4*12.8M + 6.8M + 0.8M floats ~ 236 MB
  const size_t NH = (size_t)N_NODES * HID;
  float* bufA = (float*)d_ws;           // h0, then layer-2 hl
  float* bufB = bufA + NH;              // layer-1 hl, then h1 (layer-2 input)
  float* bufC = bufB + NH;              // layer-1 hr, layer-2 hr, then h2
  float* agg = bufC + NH;               // aggregation target (both layers)
  float* logits = agg + NH;             // [ETOT, HEADS] logits -> exp weights
  float* mbuf = logits + (size_t)ETOT * HEADS;  // [N, HEADS] segment max
  float* denom = mbuf + (size_t)N_NODES * HEADS;

  const int rowTiles = N_NODES / 16;                    // 6250
  const int edgeBlocks = (ETOT + 7) / 8;                // 8 warps/block
  const int ehBlocks = (ETOT * HEADS + 255) / 256;
  const int nodeBlocks = (N_NODES + 7) / 8;
  const int initBlocks = (int)((NH + 255) / 256);

  // h0 = x @ Wp + bp
  proj_kernel<<<rowTiles, 256, 0, stream>>>(x, Wp, bp, bufA);

  // ---- GATv2 layer 1: h1 = LN(gelu(gat(h0)) + h0) ----
  dual_gemm_kernel<<<rowTiles, 256, 0, stream>>>(bufA, W1l, W1r, bufB, bufC);
  init_kernel<<<initBlocks, 256, 0, stream>>>(agg, mbuf, denom);
  edge_logits_kernel<<<edgeBlocks, 256, 0, stream>>>(ei, bufB, bufC, att1,
                                                     logits, mbuf);
  edge_exp_kernel<<<ehBlocks, 256, 0, stream>>>(ei, logits, mbuf, denom);
  edge_scatter_kernel<<<edgeBlocks, 256, 0, stream>>>(ei, bufB, logits, denom,
                                                      agg);
  post_kernel<<<nodeBlocks, 256, 0, stream>>>(agg, b1, bufA, g1, be1, bufB);

  // ---- GATv2 layer 2: h2 = LN(gelu(gat(h1)) + h1) ----
  dual_gemm_kernel<<<rowTiles, 256, 0, stream>>>(bufB, W2l, W2r, bufA, bufC);
  init_kernel<<<initBlocks, 256, 0, stream>>>(agg, mbuf, denom);
  edge_logits_kernel<<<edgeBlocks, 256, 0, stream>>>(ei, bufA, bufC, att2,
                                                     logits, mbuf);
  edge_exp_kernel<<<ehBlocks, 256, 0, stream>>>(ei, logits, mbuf, denom);
  edge_scatter_kernel<<<edgeBlocks, 256, 0, stream>>>(ei, bufA, logits, denom,
                                                      agg);
  post_kernel<<<nodeBlocks, 256, 0, stream>>>(agg, b2, bufB, g2, be2, bufC);

  // out = h2 @ Wo + bo
  out_gemm_kernel<<<rowTiles, 128, 0, stream>>>(bufC, Wo, bo, out);
}